// FlowPointContrastPad_17480516895015
// MI455X (gfx1250) — compile-verified
//
#include <hip/hip_runtime.h>
#include <hip/hip_bf16.h>

// Problem constants (reference: B=4, N=2048, D=64)
constexpr int B = 4;
constexpr int N = 2048;
constexpr int D = 64;
constexpr int NWG   = (B * N) / 16;   // 512 row-strip blocks
constexpr int WAVES = 4;              // waves per block (split over column tiles)

#define EPS_NORM 1e-7f
#define EPS_FLOW 1e-8f
#define GAMMA_C  5.0f
#define SIGMA_C  0.1f
#define TAU_FLOW_C 0.8f
#define TAU_COLOR_C 0.7f
#define TAU_PROX_C 0.5f
#define INV_SQRT3 0.57735026918962576f

typedef __attribute__((ext_vector_type(2))) float v2f;
typedef __attribute__((ext_vector_type(8))) float v8f;

// ---------------------------------------------------------------------------
// init: zero the per-batch valid-count accumulators
// ---------------------------------------------------------------------------
__global__ void fp_init_kernel(float* valid) {
    if (threadIdx.x < B) valid[threadIdx.x] = 0.0f;
}

// ---------------------------------------------------------------------------
// setup: one wave (32 lanes) per point.
//  - normalize 64-d feature row, store in WMMA-packed layout:
//      packed[point][64] : floats [h*32 + 2k + pos] = nf[col = 4k + 2h + pos]
//    so a lane of half-wave h reads its 16 K-step fragments as 8 float4s.
//  - pack per-point record (12 floats): flow(3), |flow|, rgb/255(3), mask,
//    p(3), label(float)
//  - accumulate valid[b] (sums of exact 1.0f -> order independent)
// ---------------------------------------------------------------------------
__global__ void fp_setup_kernel(const float* __restrict__ feat,
                                const float* __restrict__ flow,
                                const float* __restrict__ pts,
                                const int*   __restrict__ colors,
                                const int*   __restrict__ sam,
                                const unsigned char* __restrict__ mask,
                                float* __restrict__ nfp,
                                float* __restrict__ pt,
                                float* __restrict__ valid) {
    int pid  = blockIdx.x * (blockDim.x >> 5) + (threadIdx.x >> 5);
    int lane = threadIdx.x & 31;
    if (pid >= B * N) return;

    const float* f = feat + (size_t)pid * D;
    float x0 = f[lane];
    float x1 = f[lane + 32];
    float ss = x0 * x0 + x1 * x1;
    #pragma unroll
    for (int m = 16; m >= 1; m >>= 1) ss += __shfl_xor(ss, m, 32);
    float inv = 1.0f / (sqrtf(ss) + EPS_NORM);

    // col -> packed index: h=(c>>1)&1, k=c>>2, pos=c&1 -> h*32 + 2k + pos
    int idx0 = ((lane >> 1) & 1) * 32 + (lane >> 2) * 2 + (lane & 1);
    float* o = nfp + (size_t)pid * D;
    o[idx0]      = x0 * inv;   // col = lane
    o[idx0 + 16] = x1 * inv;   // col = lane + 32 (same h/pos, k += 8)

    if (lane == 0) {
        float fx = flow[3 * pid + 0], fy = flow[3 * pid + 1], fz = flow[3 * pid + 2];
        float fn = sqrtf(fx * fx + fy * fy + fz * fz);
        float mk = mask[pid] ? 1.0f : 0.0f;
        float* P = pt + (size_t)pid * 12;
        P[0] = fx; P[1] = fy; P[2] = fz; P[3] = fn;
        P[4] = (float)colors[3 * pid + 0] * (1.0f / 255.0f);
        P[5] = (float)colors[3 * pid + 1] * (1.0f / 255.0f);
        P[6] = (float)colors[3 * pid + 2] * (1.0f / 255.0f);
        P[7] = mk;
        P[8]  = pts[3 * pid + 0];
        P[9]  = pts[3 * pid + 1];
        P[10] = pts[3 * pid + 2];
        P[11] = (float)sam[pid];
        atomicAdd(&valid[pid / N], mk);
    }
}

__device__ __forceinline__ void fp_accum(float s, float tau, float f,
                                         float& sp, float& sm) {
    float e  = __expf(GAMMA_C * (tau - s));
    float gp = 1.0f / (1.0f + e);
    float gm = 1.0f - gp;
    sp += __expf(f * gp);
    sm += __expf(-f * gm);
}

// Load 16 B fragments (8 clause-able b128 loads), then chain 16 WMMAs.
__device__ __forceinline__ v8f fp_gram_tile(const float* __restrict__ bbase,
                                            const v2f* __restrict__ afrag) {
    v2f bfrag[16];
    #pragma unroll
    for (int j = 0; j < 8; ++j) {
        float4 t = *(const float4*)(bbase + 4 * j);
        bfrag[2 * j].x     = t.x; bfrag[2 * j].y     = t.y;
        bfrag[2 * j + 1].x = t.z; bfrag[2 * j + 1].y = t.w;
    }
    v8f c = {};
    #pragma unroll
    for (int k = 0; k < 16; ++k) {
        c = __builtin_amdgcn_wmma_f32_16x16x4_f32(
                false, afrag[k], false, bfrag[k], (short)0, c, false, false);
    }
    return c;
}

// ---------------------------------------------------------------------------
// main fused kernel: 4 waves per block, one 16-row strip per block.
// Wave w handles column tiles m0 = (w + 4*i)*16.  Pass 1 accumulates the
// sigmoid-exp sums + SAM neg-sum; cross-wave combine in LDS; pass 2 recomputes
// the Gram tiles and accumulates the SAM positive term; fixed-order finish.
// ---------------------------------------------------------------------------
__global__ __launch_bounds__(32 * WAVES)
void fp_tile_kernel(const float* __restrict__ nfp,
                    const float* __restrict__ pt,
                    const float* __restrict__ valid,
                    float* __restrict__ partials) {
    const int wg   = blockIdx.x;              // 0 .. NWG-1
    const int b    = wg / (N / 16);
    const int r0   = (wg % (N / 16)) * 16;
    const int w    = threadIdx.x >> 5;        // wave in block
    const int lane = threadIdx.x & 31;
    const int l16  = lane & 15;
    const int half = lane >> 4;

    __shared__ float rowdat[16 * 12];
    __shared__ float accbuf[WAVES][16][8];    // [wave][row][sp0,sm0,sp1,sm1,sp2,sm2,neg,pcnt]
    __shared__ float postbuf[WAVES][16];
    __shared__ float rowsum[16];

    {
        const float* ptb = pt + ((size_t)b * N + r0) * 12;
        for (int i = threadIdx.x; i < 16 * 12; i += 32 * WAVES) rowdat[i] = ptb[i];
    }
    __syncthreads();

    const float* nfb = nfp + (size_t)b * N * D;

    // preload A fragments for rows r0..r0+15 (packed: 8 x b128)
    const float* abase = nfb + (size_t)(r0 + l16) * D + half * 32;
    v2f afrag[16];
    #pragma unroll
    for (int j = 0; j < 8; ++j) {
        float4 t = *(const float4*)(abase + 4 * j);
        afrag[2 * j].x     = t.x; afrag[2 * j].y     = t.y;
        afrag[2 * j + 1].x = t.z; afrag[2 * j + 1].y = t.w;
    }

    float sp[3][8], sm[3][8], negs[8], pcnt[8];
    #pragma unroll
    for (int r = 0; r < 8; ++r) {
        sp[0][r] = sm[0][r] = sp[1][r] = sm[1][r] = sp[2][r] = sm[2][r] = 0.0f;
        negs[r] = 0.0f; pcnt[r] = 0.0f;
    }

    // ---------------- pass 1 ----------------
    for (int m0 = w * 16; m0 < N; m0 += 16 * WAVES) {
        v8f c = fp_gram_tile(nfb + (size_t)(m0 + l16) * D + half * 32, afrag);

        const float* cp = pt + ((size_t)b * N + m0 + l16) * 12;
        float4 cf = *(const float4*)(cp + 0);   // flow.xyz, |flow|
        float4 cc = *(const float4*)(cp + 4);   // rgb/255, mask
        float4 cq = *(const float4*)(cp + 8);   // point.xyz, label

        #pragma unroll
        for (int r = 0; r < 8; ++r) {
            const float* R = &rowdat[(r + 8 * half) * 12];
            float f  = c[r];
            float mo = R[7] * cc.w;

            float s_flow = (R[0] * cf.x + R[1] * cf.y + R[2] * cf.z) /
                           (R[3] * cf.w + EPS_FLOW) * mo;
            float dx = R[4] - cc.x, dy = R[5] - cc.y, dz = R[6] - cc.z;
            float s_col = (1.0f - sqrtf(dx * dx + dy * dy + dz * dz) * INV_SQRT3) * mo;
            float px = R[8] - cq.x, py = R[9] - cq.y, pz = R[10] - cq.z;
            float s_prox = __expf(-sqrtf(px * px + py * py + pz * pz) *
                                  (1.0f / (2.0f * SIGMA_C * SIGMA_C))) * mo;

            fp_accum(s_flow, TAU_FLOW_C,  f, sp[0][r], sm[0][r]);
            fp_accum(s_col,  TAU_COLOR_C, f, sp[1][r], sm[1][r]);
            fp_accum(s_prox, TAU_PROX_C,  f, sp[2][r], sm[2][r]);

            float ex = __expf(f);
            bool ispos = (R[11] == cq.w);
            negs[r] += ispos ? 0.0f : ex;
            pcnt[r] += ispos ? 1.0f : 0.0f;
        }
    }

    // in-wave reduction across each 16-lane half
    #pragma unroll
    for (int r = 0; r < 8; ++r) {
        #pragma unroll
        for (int m = 1; m < 16; m <<= 1) {
            sp[0][r] += __shfl_xor(sp[0][r], m, 32);
            sm[0][r] += __shfl_xor(sm[0][r], m, 32);
            sp[1][r] += __shfl_xor(sp[1][r], m, 32);
            sm[1][r] += __shfl_xor(sm[1][r], m, 32);
            sp[2][r] += __shfl_xor(sp[2][r], m, 32);
            sm[2][r] += __shfl_xor(sm[2][r], m, 32);
            negs[r]  += __shfl_xor(negs[r],  m, 32);
            pcnt[r]  += __shfl_xor(pcnt[r],  m, 32);
        }
    }

    if (l16 == 0) {
        #pragma unroll
        for (int r = 0; r < 8; ++r) {
            int row = r + 8 * half;
            accbuf[w][row][0] = sp[0][r]; accbuf[w][row][1] = sm[0][r];
            accbuf[w][row][2] = sp[1][r]; accbuf[w][row][3] = sm[1][r];
            accbuf[w][row][4] = sp[2][r]; accbuf[w][row][5] = sm[2][r];
            accbuf[w][row][6] = negs[r];  accbuf[w][row][7] = pcnt[r];
        }
    }
    __syncthreads();

    // complete per-row SAM neg-sums (broadcast reads, fixed order)
    float negt[8];
    #pragma unroll
    for (int r = 0; r < 8; ++r) {
        int row = r + 8 * half;
        float s = 0.0f;
        #pragma unroll
        for (int ww = 0; ww < WAVES; ++ww) s += accbuf[ww][row][6];
        negt[r] = s;
    }

    // ---------------- pass 2: SAM positive term ----------------
    float post[8];
    #pragma unroll
    for (int r = 0; r < 8; ++r) post[r] = 0.0f;

    for (int m0 = w * 16; m0 < N; m0 += 16 * WAVES) {
        v8f c = fp_gram_tile(nfb + (size_t)(m0 + l16) * D + half * 32, afrag);
        float labcol = pt[((size_t)b * N + m0 + l16) * 12 + 11];
        #pragma unroll
        for (int r = 0; r < 8; ++r) {
            float labrow = rowdat[(r + 8 * half) * 12 + 11];
            float f = c[r];
            float term = __logf(__expf(f) + negt[r]) - f;
            post[r] += (labrow == labcol) ? term : 0.0f;
        }
    }

    #pragma unroll
    for (int r = 0; r < 8; ++r) {
        #pragma unroll
        for (int m = 1; m < 16; m <<= 1) post[r] += __shfl_xor(post[r], m, 32);
    }
    if (l16 == 0) {
        #pragma unroll
        for (int r = 0; r < 8; ++r) postbuf[w][r + 8 * half] = post[r];
    }
    __syncthreads();

    // fixed-order per-row finish (threads 0..15, one row each)
    if (threadIdx.x < 16) {
        int row = threadIdx.x;
        float q[8];
        #pragma unroll
        for (int qi = 0; qi < 8; ++qi) {
            float s = 0.0f;
            #pragma unroll
            for (int ww = 0; ww < WAVES; ++ww) s += accbuf[ww][row][qi];
            q[qi] = s;
        }
        float pterm = 0.0f;
        #pragma unroll
        for (int ww = 0; ww < WAVES; ++ww) pterm += postbuf[ww][row];

        float vb   = valid[b];
        float invB = 1.0f / (float)B;
        float mrow = rowdat[row * 12 + 7];
        float within = log1pf(q[0]) + log1pf(q[1]) + log1pf(q[2]) +
                       log1pf(q[3]) + log1pf(q[4]) + log1pf(q[5]);
        float samr = pterm / q[7];                 // pcnt >= 1 (diagonal)
        rowsum[row] = -invB * mrow * within / vb
                    +  invB * samr * (1.0f / (float)N);
    }
    __syncthreads();
    if (threadIdx.x == 0) {
        float s = 0.0f;
        #pragma unroll
        for (int row = 0; row < 16; ++row) s += rowsum[row];
        partials[wg] = s;
    }
}

// ---------------------------------------------------------------------------
// deterministic final reduction of the 512 per-block partials
// ---------------------------------------------------------------------------
__global__ void fp_reduce_kernel(const float* __restrict__ partials,
                                 float* __restrict__ out, int n) {
    __shared__ float s[512];
    int t = threadIdx.x;
    s[t] = (t < n) ? partials[t] : 0.0f;
    __syncthreads();
    for (int w = 256; w >= 1; w >>= 1) {
        if (t < w) s[t] += s[t + w];
        __syncthreads();
    }
    if (t == 0) out[0] = s[0];
}

// ---------------------------------------------------------------------------
extern "C" void kernel_launch(void* const* d_in, const int* in_sizes, int n_in,
                              void* d_out, int out_size, void* d_ws, size_t ws_size,
                              hipStream_t stream) {
    const float* feat = (const float*)d_in[0];
    const float* flow = (const float*)d_in[1];
    const float* pts  = (const float*)d_in[2];
    const int*   col  = (const int*)d_in[3];
    const int*   sam  = (const int*)d_in[4];
    const unsigned char* mask = (const unsigned char*)d_in[5];
    float* out = (float*)d_out;

    float* nfp      = (float*)d_ws;                       // B*N*D (packed)
    float* pt       = nfp + (size_t)B * N * D;            // B*N*12
    float* valid    = pt + (size_t)B * N * 12;            // B (pad 16)
    float* partials = valid + 16;                         // NWG

    fp_init_kernel<<<1, 32, 0, stream>>>(valid);
    fp_setup_kernel<<<(B * N + 7) / 8, 256, 0, stream>>>(
        feat, flow, pts, col, sam, mask, nfp, pt, valid);
    fp_tile_kernel<<<NWG, 32 * WAVES, 0, stream>>>(nfp, pt, valid, partials);
    fp_reduce_kernel<<<1, 512, 0, stream>>>(partials, out, NWG);
}